// SpatialConvLayer_85933705658979
// MI455X (gfx1250) — compile-verified
//
#include <hip/hip_runtime.h>
#include <hip/hip_bf16.h>

// Problem constants (match reference).
#define BB 8
#define NN 10000
#define CIN 16
#define TT 24
#define COUT 32
#define EE 160000
#define BT (BB * TT)          // 192
#define RR (NN * BT)          // 1,920,000 rows

typedef float v2f __attribute__((ext_vector_type(2)));
typedef float v8f __attribute__((ext_vector_type(8)));

// ---------------------------------------------------------------------------
// 1) Pack (B,N,C,T) f32 -> X0 [n][bt][c]  (bt = b*T + t)
// ---------------------------------------------------------------------------
__global__ void __launch_bounds__(256) pack_kernel(const float* __restrict__ xin,
                                                   float* __restrict__ X0) {
    int idx = blockIdx.x * 256 + threadIdx.x;        // < RR*CIN = 30,720,000
    int c  = idx & 15;
    int bt = (idx >> 4) % BT;
    int n  = idx / (16 * BT);
    int b  = bt / TT;
    int t  = bt - b * TT;
    X0[idx] = xin[(((size_t)b * NN + n) * CIN + c) * TT + t];
}

// ---------------------------------------------------------------------------
// 2) CSR build: zero counters, histogram of dst, exclusive scan, fill columns
// ---------------------------------------------------------------------------
__global__ void __launch_bounds__(256) zero_kernel(int* __restrict__ p, int n) {
    int i = blockIdx.x * 256 + threadIdx.x;
    if (i < n) p[i] = 0;
}

__global__ void __launch_bounds__(256) hist_kernel(const int* __restrict__ ei,
                                                   int* __restrict__ cnt) {
    int e = blockIdx.x * 256 + threadIdx.x;          // E = 160000 = 625*256
    int d = ei[EE + e];                              // dst row of edge_index
    atomicAdd(&cnt[d], 1);
}

__global__ void __launch_bounds__(1024) scan_kernel(const int* __restrict__ cnt,
                                                    int* __restrict__ row_ptr,
                                                    int* __restrict__ cursor) {
    __shared__ int sums[1024];
    const int tid = threadIdx.x;
    const int CH  = 10;                              // 1024*10 >= NN
    const int beg = tid * CH;
    int s = 0;
    #pragma unroll
    for (int i = 0; i < CH; ++i) {
        int idx = beg + i;
        if (idx < NN) s += cnt[idx];
    }
    sums[tid] = s;
    __syncthreads();
    for (int off = 1; off < 1024; off <<= 1) {
        int v   = sums[tid];
        int add = (tid >= off) ? sums[tid - off] : 0;
        __syncthreads();
        sums[tid] = v + add;
        __syncthreads();
    }
    int run = (tid > 0) ? sums[tid - 1] : 0;
    #pragma unroll
    for (int i = 0; i < CH; ++i) {
        int idx = beg + i;
        if (idx < NN) {
            row_ptr[idx] = run;
            cursor[idx]  = run;
            run += cnt[idx];
        }
    }
    if (tid == 0) row_ptr[NN] = sums[1023];          // == E
}

__global__ void __launch_bounds__(256) fill_kernel(const int* __restrict__ ei,
                                                   int* __restrict__ cursor,
                                                   int* __restrict__ col) {
    int e = blockIdx.x * 256 + threadIdx.x;
    int d = ei[EE + e];
    int pos = atomicAdd(&cursor[d], 1);
    col[pos] = ei[e];                                // src
}

// ---------------------------------------------------------------------------
// 3) Gather aggregation: H[n] = X[n] + sum_{j -> n} X[j], contiguous rows.
//    One block per node; fully coalesced float4 traffic, register accumulators.
// ---------------------------------------------------------------------------
template <int C>
__global__ void __launch_bounds__(256) agg_kernel(const float* __restrict__ X,
                                                  const int* __restrict__ row_ptr,
                                                  const int* __restrict__ col,
                                                  float* __restrict__ H) {
    constexpr int V4  = (BT * C) / 4;                // float4 per node (768 / 1536)
    constexpr int PER = V4 / 256;                    // 3 / 6
    const int n = blockIdx.x;
    const float4* xn = (const float4*)X + (size_t)n * V4;
    float4 acc[PER];
    #pragma unroll
    for (int i = 0; i < PER; ++i) acc[i] = xn[threadIdx.x + i * 256];
    const int pbeg = row_ptr[n], pend = row_ptr[n + 1];
    for (int p = pbeg; p < pend; ++p) {
        const float4* xs = (const float4*)X + (size_t)col[p] * V4;
        if (p + 1 < pend)  // prefetch next neighbor block (global_prefetch_b8)
            __builtin_prefetch((const float4*)X + (size_t)col[p + 1] * V4 + threadIdx.x, 0, 1);
        #pragma unroll
        for (int i = 0; i < PER; ++i) {
            float4 v = xs[threadIdx.x + i * 256];
            acc[i].x += v.x; acc[i].y += v.y; acc[i].z += v.z; acc[i].w += v.w;
        }
    }
    float4* hn = (float4*)H + (size_t)n * V4;
    #pragma unroll
    for (int i = 0; i < PER; ++i) hn[threadIdx.x + i * 256] = acc[i];
}

// ---------------------------------------------------------------------------
// WMMA f32 16x16x4 helpers. Operand layout (ISA 7.12.2):
//   A (16x4):  lane l, comp j -> M = l&15, K = 2*(l>>4) + j
//   B (4x16):  lane l, comp j -> K = 2*(l>>4) + j, N = l&15
//   C/D (16x16): lane l, vgpr v -> M = v + 8*(l>>4), N = l&15
// ---------------------------------------------------------------------------
__device__ __forceinline__ v8f wmma_f32(v2f a, v2f b, v8f c) {
    return __builtin_amdgcn_wmma_f32_16x16x4_f32(false, a, false, b, (short)0, c,
                                                 false, false);
}

// Cooperative load of a KxNCOL weight matrix into LDS, pre-swizzled into
// B-operand order so each per-lane B fetch is ONE contiguous ds_load_b64:
//   wswz[(kp*NCOL + col)*2 + j] = W[(2*kp + j)*NCOL + col]
// Lane l, k-block kb then reads float2 at (( (kb*2 + (l>>4))*NCOL + col )*2).
template <int K, int NCOL, int NTHREADS>
__device__ __forceinline__ void load_wswz(const float* __restrict__ W,
                                          float* __restrict__ wswz) {
    for (int i = threadIdx.x; i < K * NCOL; i += NTHREADS) {
        int j   = i & 1;
        int col = (i >> 1) % NCOL;
        int kp  = i / (2 * NCOL);
        wswz[i] = W[(2 * kp + j) * NCOL + col];
    }
}

// ---------------------------------------------------------------------------
// 4) Layer-1 MLP: X1 = relu(H1 @ W1 + b1) @ W2 + b2      (16 -> 64 -> 32)
//    One wave per 16-row tile; 8 waves per block.
// ---------------------------------------------------------------------------
__global__ void __launch_bounds__(256) mlp1_kernel(const float* __restrict__ H1,
                                                   const float* __restrict__ W1,
                                                   const float* __restrict__ b1,
                                                   const float* __restrict__ W2,
                                                   const float* __restrict__ b2,
                                                   float* __restrict__ X1) {
    __shared__ alignas(16) float w1s[16 * 64];       // swizzled
    __shared__ alignas(16) float w2s[64 * 32];       // swizzled
    __shared__ alignas(16) float b1s[64];
    __shared__ alignas(16) float b2s[32];
    __shared__ alignas(16) float h1s[8][16 * 64];    // per-wave staging (32 KB)

    load_wswz<16, 64, 256>(W1, w1s);
    load_wswz<64, 32, 256>(W2, w2s);
    if (threadIdx.x < 64) b1s[threadIdx.x] = b1[threadIdx.x];
    if (threadIdx.x < 32) b2s[threadIdx.x] = b2[threadIdx.x];
    __syncthreads();

    const int lane  = threadIdx.x & 31;
    const int wid   = threadIdx.x >> 5;
    const int r0    = (blockIdx.x * 8 + wid) * 16;   // tile base row
    const int m     = lane & 15;
    const int hi    = lane >> 4;                     // 0 / 1
    const int khalf = hi * 2;
    const int ncol  = lane & 15;

    // A operands for K = 16 (4 k-blocks).
    v2f a[4];
    const float* arow = H1 + (size_t)(r0 + m) * 16;
    #pragma unroll
    for (int kb = 0; kb < 4; ++kb)
        a[kb] = *(const v2f*)(arow + kb * 4 + khalf);

    float* myh = h1s[wid];

    // Stage 1: 16x16 @ 16x64, bias + relu, stage into LDS (A-swizzle layout).
    #pragma unroll
    for (int nb = 0; nb < 4; ++nb) {
        v8f c = {};
        #pragma unroll
        for (int kb = 0; kb < 4; ++kb) {
            v2f bop = *(const v2f*)(w1s + ((kb * 2 + hi) * 64 + nb * 16 + ncol) * 2);
            c = wmma_f32(a[kb], bop, c);
        }
        float bias = b1s[nb * 16 + ncol];
        #pragma unroll
        for (int v = 0; v < 8; ++v) {
            float h = c[v] + bias;
            h = h > 0.f ? h : 0.f;
            myh[(v + hi * 8) * 64 + nb * 16 + ncol] = h;
        }
    }

    // Stage 2: 16x64 @ 64x32, + bias, store (contiguous [n][bt][32] layout).
    float* outb = X1 + (size_t)r0 * 32;
    #pragma unroll
    for (int nb2 = 0; nb2 < 2; ++nb2) {
        v8f d = {};
        #pragma unroll
        for (int kb = 0; kb < 16; ++kb) {
            v2f a2 = *(const v2f*)(myh + m * 64 + kb * 4 + khalf);
            v2f bop = *(const v2f*)(w2s + ((kb * 2 + hi) * 32 + nb2 * 16 + ncol) * 2);
            d = wmma_f32(a2, bop, d);
        }
        float bias = b2s[nb2 * 16 + ncol];
        #pragma unroll
        for (int v = 0; v < 8; ++v)
            outb[(v + hi * 8) * 32 + nb2 * 16 + ncol] = d[v] + bias;
    }
}

// ---------------------------------------------------------------------------
// 5) Layer-2 MLP + final relu + transpose store to (B,N,C_OUT,T).
//    out = relu( relu(H2 @ W3 + b3) @ W4 + b4 )        (32 -> 128 -> 32)
//    One wave per 16-row tile; 4 waves per block (LDS budget).
// ---------------------------------------------------------------------------
__global__ void __launch_bounds__(128) mlp2_kernel(const float* __restrict__ H2,
                                                   const float* __restrict__ W3,
                                                   const float* __restrict__ b3,
                                                   const float* __restrict__ W4,
                                                   const float* __restrict__ b4,
                                                   float* __restrict__ out) {
    __shared__ alignas(16) float w3s[32 * 128];      // swizzled
    __shared__ alignas(16) float w4s[128 * 32];      // swizzled
    __shared__ alignas(16) float b3s[128];
    __shared__ alignas(16) float b4s[32];
    __shared__ alignas(16) float h3s[4][16 * 128];   // per-wave staging (32 KB)

    load_wswz<32, 128, 128>(W3, w3s);
    load_wswz<128, 32, 128>(W4, w4s);
    if (threadIdx.x < 128) b3s[threadIdx.x] = b3[threadIdx.x];
    if (threadIdx.x < 32)  b4s[threadIdx.x] = b4[threadIdx.x];
    __syncthreads();

    const int lane  = threadIdx.x & 31;
    const int wid   = threadIdx.x >> 5;
    const int r0    = (blockIdx.x * 4 + wid) * 16;
    const int m     = lane & 15;
    const int hi    = lane >> 4;
    const int khalf = hi * 2;
    const int ncol  = lane & 15;
    const int node  = r0 / BT;                       // tile never crosses a node
    const int bt0   = r0 % BT;

    // A operands for K = 32 (8 k-blocks).
    v2f a[8];
    const float* arow = H2 + (size_t)(r0 + m) * 32;
    #pragma unroll
    for (int kb = 0; kb < 8; ++kb)
        a[kb] = *(const v2f*)(arow + kb * 4 + khalf);

    float* myh = h3s[wid];

    // Stage 1: 16x32 @ 32x128, bias + relu, stage into LDS (A-swizzle layout).
    #pragma unroll
    for (int nb = 0; nb < 8; ++nb) {
        v8f c = {};
        #pragma unroll
        for (int kb = 0; kb < 8; ++kb) {
            v2f bop = *(const v2f*)(w3s + ((kb * 2 + hi) * 128 + nb * 16 + ncol) * 2);
            c = wmma_f32(a[kb], bop, c);
        }
        float bias = b3s[nb * 16 + ncol];
        #pragma unroll
        for (int v = 0; v < 8; ++v) {
            float h = c[v] + bias;
            h = h > 0.f ? h : 0.f;
            myh[(v + hi * 8) * 128 + nb * 16 + ncol] = h;
        }
    }

    // Stage 2: 16x128 @ 128x32, bias + final relu, transposed store.
    #pragma unroll
    for (int nb2 = 0; nb2 < 2; ++nb2) {
        v8f d = {};
        #pragma unroll
        for (int kb = 0; kb < 32; ++kb) {
            v2f a2 = *(const v2f*)(myh + m * 128 + kb * 4 + khalf);
            v2f bop = *(const v2f*)(w4s + ((kb * 2 + hi) * 32 + nb2 * 16 + ncol) * 2);
            d = wmma_f32(a2, bop, d);
        }
        const int co = nb2 * 16 + ncol;
        float bias = b4s[co];
        #pragma unroll
        for (int v = 0; v < 8; ++v) {
            float h = d[v] + bias;
            h = h > 0.f ? h : 0.f;
            int bt = bt0 + v + hi * 8;
            int b  = bt / TT;
            int t  = bt - b * TT;
            out[(((size_t)b * NN + node) * COUT + co) * TT + t] = h;
        }
    }
}

// ---------------------------------------------------------------------------
// Host-side orchestration. Workspace layout (bytes):
//   [0,              245,760,000)  region A: X0 (R*16 f32) + H1 (R*16 f32);
//                                  reused whole as H2 (R*32 f32) in phase 2
//   [245,760,000,    491,520,000)  X1 (R*32 f32)
//   [491,520,000, ...)             CSR: row_ptr, cursor, cnt, col
// ---------------------------------------------------------------------------
extern "C" void kernel_launch(void* const* d_in, const int* in_sizes, int n_in,
                              void* d_out, int out_size, void* d_ws, size_t ws_size,
                              hipStream_t stream) {
    const float* xin = (const float*)d_in[0];
    const int*   ei  = (const int*)d_in[1];
    const float* W1  = (const float*)d_in[2];
    const float* b1  = (const float*)d_in[3];
    const float* W2  = (const float*)d_in[4];
    const float* b2  = (const float*)d_in[5];
    const float* W3  = (const float*)d_in[6];
    const float* b3  = (const float*)d_in[7];
    const float* W4  = (const float*)d_in[8];
    const float* b4  = (const float*)d_in[9];
    float* out = (float*)d_out;

    char* ws = (char*)d_ws;
    const size_t szA = (size_t)RR * 32 * sizeof(float);      // 245,760,000
    float* X0 = (float*)ws;                                  // R*16
    float* H1 = X0 + (size_t)RR * 16;                        // R*16
    float* X1 = (float*)(ws + szA);                          // R*32
    float* H2 = (float*)ws;                                  // R*32 (phase 2)
    int* row_ptr = (int*)(ws + 2 * szA);
    int* cursor  = row_ptr + (NN + 16);
    int* cnt     = cursor + NN;
    int* col     = cnt + NN;

    // Phase 0: pack + CSR build.
    pack_kernel<<<(RR * CIN) / 256, 256, 0, stream>>>(xin, X0);
    zero_kernel<<<(NN + 255) / 256, 256, 0, stream>>>(cnt, NN);
    hist_kernel<<<EE / 256, 256, 0, stream>>>(ei, cnt);
    scan_kernel<<<1, 1024, 0, stream>>>(cnt, row_ptr, cursor);
    fill_kernel<<<EE / 256, 256, 0, stream>>>(ei, cursor, col);

    // Layer 1.
    agg_kernel<CIN><<<NN, 256, 0, stream>>>(X0, row_ptr, col, H1);
    mlp1_kernel<<<(RR / 16) / 8, 256, 0, stream>>>(H1, W1, b1, W2, b2, X1);

    // Layer 2 (H2 reuses region A; X0/H1 are dead).
    agg_kernel<COUT><<<NN, 256, 0, stream>>>(X1, row_ptr, col, H2);
    mlp2_kernel<<<(RR / 16) / 4, 128, 0, stream>>>(H2, W3, b3, W4, b4, out);
}